// CatmullRomSplineMotion_53712861004510
// MI455X (gfx1250) — compile-verified
//
#include <hip/hip_runtime.h>
#include <math.h>

// ---------------------------------------------------------------------------
// Catmull-Rom spline evaluation, MI455X (gfx1250).
//   d_in[0]: cps  (1000 x 2 f32)
//   d_in[1]: tq   (8388608 f32)
//   d_out  : out  (8388608 x 2 f32)
//   d_ws   : staging: aux (float2 x1003) | knots (f32 x1003) | LUT (u16 x4096)
// Eval blocks async-stage the 20 KB table region into LDS via CDNA5
// global->LDS async DMA (global_load_async_to_lds_b128 + s_wait_asynccnt),
// then evaluate 4096 queries each: uniform-bucket LUT + short scan replaces
// a 10-step binary search. HBM traffic floor: 12 B/query (~100 MB total).
// ---------------------------------------------------------------------------

#define CR_EPS 1e-07f
#define CR_CPN 1000
#define CR_NAUX 1003            // aux points (float2)
#define CR_NSEG 1002
#define CR_NKNOT 1003
#define CR_KNOT_OFF 2048        // float index of knots inside ws / LDS buffer
#define CR_LUT_BYTE_OFF 12288   // byte offset of u16 LUT
#define CR_LUT_N 4096
#define CR_STAGE_BYTES 20480    // bytes staged per block (5 x b128 per thread)
#define CR_QPB 4096             // queries per block (16 per thread)

// ---------------- setup: replicate build_spline + build LUT ----------------
__global__ __launch_bounds__(256) void cr_setup_kernel(
    const float* __restrict__ cps, float* __restrict__ ws) {
  __shared__ __align__(16) float2 saux[CR_NAUX];
  __shared__ float sseg[CR_NSEG];
  __shared__ float sknots[CR_NKNOT];
  const float2* cp2 = (const float2*)cps;
  int t = threadIdx.x;

  // closed loop: aux[1..1001] = [cps[0..999], cps[0]]
  for (int i = t; i < CR_CPN + 1; i += 256) {
    saux[i + 1] = (i < CR_CPN) ? cp2[i] : cp2[0];
  }
  __syncthreads();

  if (t == 0) {
    float2 c0 = cp2[0];
    float2 c1 = cp2[1];
    float2 cl = cp2[CR_CPN - 1];           // cps[999]
    float dx01 = c0.x - c1.x, dy01 = c0.y - c1.y;
    float l01 = sqrtf(dx01 * dx01 + dy01 * dy01 + CR_EPS);
    // cps[-1] = cps[0] (closed), cps[-2] = cps[999]
    float dxl = c0.x - cl.x, dyl = c0.y - cl.y;
    float l_last = sqrtf(dxl * dxl + dyl * dyl + CR_EPS);
    float r0 = l01 / l_last;
    float rl = l_last / l01;
    float2 first; first.x = c0.x - r0 * (c0.x - cl.x);
                  first.y = c0.y - r0 * (c0.y - cl.y);
    float2 last;  last.x  = c0.x + rl * (c1.x - c0.x);
                  last.y  = c0.y + rl * (c1.y - c0.y);
    saux[0] = first;
    saux[CR_NAUX - 1] = last;
  }
  __syncthreads();

  // write aux points out
  float2* waux = (float2*)ws;
  for (int i = t; i < CR_NAUX; i += 256) waux[i] = saux[i];

  // segment "lengths": (|d|^2)^(alpha/2), alpha=0.5 -> d2^0.25
  for (int i = t; i < CR_NSEG; i += 256) {
    float dx = saux[i + 1].x - saux[i].x;
    float dy = saux[i + 1].y - saux[i].y;
    float d2 = dx * dx + dy * dy;
    sseg[i] = sqrtf(sqrtf(d2));
  }
  __syncthreads();

  // serial prefix sum (1002 adds, once per launch) -> knots
  if (t == 0) {
    float acc = 0.0f;
    sknots[0] = 0.0f;
    for (int i = 0; i < CR_NSEG; ++i) {
      acc += sseg[i];
      sknots[i + 1] = acc;
    }
  }
  __syncthreads();

  // copy knots out
  for (int i = t; i < CR_NKNOT; i += 256) ws[CR_KNOT_OFF + i] = sknots[i];

  // uniform-bucket LUT over [0, T]: LUT[b] = largest idx with
  // knots[idx] <= b * (T/4096). Eval uses it as a start hint only.
  {
    float W = sknots[CR_NKNOT - 1] * (1.0f / (float)CR_LUT_N);
    unsigned short* wlut = (unsigned short*)((char*)ws + CR_LUT_BYTE_OFF);
    for (int b = t; b < CR_LUT_N; b += 256) {
      float target = (float)b * W;
      int lo = 0, hi = CR_NKNOT;  // virtual knots[1003] = +inf
      while (hi - lo > 1) {
        int mid = (lo + hi) >> 1;
        bool c = sknots[mid] <= target;
        lo = c ? mid : lo;
        hi = c ? hi : mid;
      }
      wlut[b] = (unsigned short)lo;
    }
  }
}

// ---------------- eval ------------------------------------------------------
__device__ __forceinline__ float2 cr_eval(const float* __restrict__ kn,
                                          const float2* __restrict__ aux,
                                          const unsigned short* __restrict__ lut,
                                          float scale, float tv) {
  // Bucket hint, then scan to the true interval:
  //   result: last idx with kn[idx] <= tv (== searchsorted(...,'right') - 1).
  // Guards are unnecessary by invariant: kn[0]=0 < tv bounds the down-scan,
  // tv < kn[top] (up to ulps, absorbed by sg clamp) bounds the up-scan.
  int b = (int)(tv * scale);
  b = b < 0 ? 0 : (b > CR_LUT_N - 1 ? CR_LUT_N - 1 : b);
  int idx = (int)lut[b];
  while (kn[idx] > tv) --idx;          // rare rounding fixup
  while (kn[idx + 1] <= tv) ++idx;     // ~1-2 steps
  int sg = idx;
  sg = sg < 1 ? 1 : sg;
  sg = sg > CR_CPN ? CR_CPN : sg;

  float t0 = kn[sg - 1], t1 = kn[sg], t2 = kn[sg + 1], t3 = kn[sg + 2];
  float2 P0 = aux[sg - 1], P1 = aux[sg], P2 = aux[sg + 1], P3 = aux[sg + 2];

  float i01 = __builtin_amdgcn_rcpf(t1 - t0);
  float i12 = __builtin_amdgcn_rcpf(t2 - t1);
  float i23 = __builtin_amdgcn_rcpf(t3 - t2);
  float i02 = __builtin_amdgcn_rcpf(t2 - t0);
  float i13 = __builtin_amdgcn_rcpf(t3 - t1);

  float a, b2;
  a = (t1 - tv) * i01; b2 = (tv - t0) * i01;
  float2 x01;  x01.x  = a * P0.x + b2 * P1.x;   x01.y  = a * P0.y + b2 * P1.y;
  float a12 = (t2 - tv) * i12, b12 = (tv - t1) * i12;
  float2 x12;  x12.x  = a12 * P1.x + b12 * P2.x; x12.y = a12 * P1.y + b12 * P2.y;
  a = (t3 - tv) * i23; b2 = (tv - t2) * i23;
  float2 x23;  x23.x  = a * P2.x + b2 * P3.x;   x23.y  = a * P2.y + b2 * P3.y;
  a = (t2 - tv) * i02; b2 = (tv - t0) * i02;
  float2 x012; x012.x = a * x01.x + b2 * x12.x; x012.y = a * x01.y + b2 * x12.y;
  a = (t3 - tv) * i13; b2 = (tv - t1) * i13;
  float2 x123; x123.x = a * x12.x + b2 * x23.x; x123.y = a * x12.y + b2 * x23.y;
  float2 r;    r.x = a12 * x012.x + b12 * x123.x;
               r.y = a12 * x012.y + b12 * x123.y;
  return r;
}

__global__ __launch_bounds__(256) void cr_eval_kernel(
    const float* __restrict__ ws, const float* __restrict__ tq,
    float2* __restrict__ out, int n) {
  __shared__ __align__(16) char sbuf[CR_STAGE_BYTES];

  // --- CDNA5 async global -> LDS staging of the spline tables (20 KB) ---
  // GVS mode: each lane copies 16 B from saddr + vaddr to LDS address vdst; 5 x b128 per thread.
  unsigned lds_base = (unsigned)(unsigned long long)&sbuf[0];
  unsigned long long wsa = (unsigned long long)ws;
#pragma unroll
  for (int k = 0; k < 5; ++k) {
    unsigned off = (threadIdx.x + k * 256u) * 16u;
    unsigned ldsa = lds_base + off;
    asm volatile("global_load_async_to_lds_b128 %0, %1, %2"
                 :
                 : "v"(ldsa), "v"(off), "s"(wsa)
                 : "memory");
  }
  asm volatile("s_wait_asynccnt 0" ::: "memory");
  __syncthreads();

  const float2* aux = (const float2*)&sbuf[0];
  const float* kn = (const float*)&sbuf[0] + CR_KNOT_OFF;
  const unsigned short* lut = (const unsigned short*)&sbuf[CR_LUT_BYTE_OFF];
  float scale = (float)CR_LUT_N / kn[CR_NKNOT - 1];

  long long base = (long long)blockIdx.x * CR_QPB + (long long)threadIdx.x * 4ll;
  // Keep rolled: the 4 independent evals inside one iteration already give
  // enough ILP to cover LDS latency; unrolling only bloats I-cache.
#pragma unroll 1
  for (int j = 0; j < 4; ++j) {
    long long i0 = base + (long long)j * 1024ll;
    if (i0 + 3 < n) {
      // fast path: one b128 load of tq, two b128 stores of output
      float4 t4 = *(const float4*)(tq + i0);
      float2 r0 = cr_eval(kn, aux, lut, scale, t4.x);
      float2 r1 = cr_eval(kn, aux, lut, scale, t4.y);
      float2 r2 = cr_eval(kn, aux, lut, scale, t4.z);
      float2 r3 = cr_eval(kn, aux, lut, scale, t4.w);
      float4* o4 = (float4*)(out + i0);
      float4 lo4; lo4.x = r0.x; lo4.y = r0.y; lo4.z = r1.x; lo4.w = r1.y;
      float4 hi4; hi4.x = r2.x; hi4.y = r2.y; hi4.z = r3.x; hi4.w = r3.y;
      o4[0] = lo4;
      o4[1] = hi4;
    } else {
      for (int jj = 0; jj < 4; ++jj) {
        long long i = i0 + jj;
        if (i < n) out[i] = cr_eval(kn, aux, lut, scale, tq[i]);
      }
    }
  }
}

// ---------------- launch ----------------------------------------------------
extern "C" void kernel_launch(void* const* d_in, const int* in_sizes, int n_in,
                              void* d_out, int out_size, void* d_ws, size_t ws_size,
                              hipStream_t stream) {
  const float* cps = (const float*)d_in[0];
  const float* tq = (const float*)d_in[1];
  float* out = (float*)d_out;
  float* ws = (float*)d_ws;
  int n = in_sizes[1];  // number of queries

  cr_setup_kernel<<<1, 256, 0, stream>>>(cps, ws);

  int blocks = (n + CR_QPB - 1) / CR_QPB;
  cr_eval_kernel<<<blocks, 256, 0, stream>>>(ws, tq, (float2*)out, n);
}